// Ga2MaxPool2d_38096359915956
// MI455X (gfx1250) — compile-verified
//
#include <hip/hip_runtime.h>
#include <stdint.h>

// GA2 max-pool, K=2, stride 2, fp32.
// x: (B=16, C4=256, H=128, W=128) -> out: (16, 256, 64, 64)
// Channels grouped in quads (4 GA2 components). Pick the window position with
// the largest sum-of-squares across the 4 components (row-major first-match),
// emit all 4 components at that position.
//
// Memory-bound (~320 MiB @ 23.3 TB/s ~= 14 us). CDNA5 path: stage tiles into
// LDS with global_load_async_to_lds_b128 (ASYNCcnt), fence with
// s_wait_asynccnt, compute from LDS with ds_load_b64.

#define G_TOTAL      1024   // B * C = 16 * 64 component-quads
#define PLANE        16384  // H * W floats
#define QUAD_IN      65536  // 4 * PLANE
#define OUT_PLANE    4096   // Ho * Wo
#define QUAD_OUT     16384  // 4 * OUT_PLANE
#define HB_PER_G     16     // Ho / 4 row-blocks per quad
#define W_IN         128
#define WO           64

__device__ __forceinline__ void async_copy_b128(uint32_t lds_byte_off,
                                                const float* gptr) {
  // Per-lane: copies 16 bytes from the global address gptr to the LDS byte offset lds_byte_off. Tracked by ASYNCcnt.
  asm volatile("global_load_async_to_lds_b128 %0, %1, off"
               :
               : "v"(lds_byte_off), "v"((unsigned long long)(uintptr_t)gptr)
               : "memory");
}

__device__ __forceinline__ void wait_asynccnt0() {
  asm volatile("s_wait_asynccnt 0" ::: "memory");
}

__global__ __launch_bounds__(256) void ga2_maxpool_kernel(
    const float* __restrict__ x, float* __restrict__ out) {
  __shared__ __align__(16) float smem[4 * 1024];  // 4 comps x (8 rows x 128) = 16KB

  const int tid = threadIdx.x;           // 0..255
  const int g   = blockIdx.x >> 4;       // component-quad id (0..1023)
  const int hb  = blockIdx.x & 15;       // 4-output-row block (0..15)

  // LDS byte offset of our shared buffer within the workgroup allocation:
  // generic LDS pointers carry the offset in their low 32 bits.
  const uint32_t lds_base = (uint32_t)(uintptr_t)(void*)smem;

  // ---- Stage 16KB: component k slab = 8 contiguous input rows (4KB each). ----
  // Global floats: g*QUAD_IN + k*PLANE + hb*1024 + tid*4  (16B per lane).
  const float* gbase = x + (size_t)g * QUAD_IN + (size_t)hb * (8 * W_IN) + tid * 4;
#pragma unroll
  for (int k = 0; k < 4; ++k) {
    async_copy_b128(lds_base + (uint32_t)(k * 4096 + tid * 16),
                    gbase + (size_t)k * PLANE);
  }

  wait_asynccnt0();   // our wave's LDS writes are complete
  __syncthreads();    // make them visible across the workgroup

  // ---- Compute: thread -> (local output row r, output col wo). ----
  const int r  = tid >> 6;   // 0..3
  const int wo = tid & 63;   // 0..63

  float w0[4], w1[4], w2[4], w3[4];  // window positions p = dy*2+dx, per comp
#pragma unroll
  for (int k = 0; k < 4; ++k) {
    const int base = k * 1024 + (2 * r) * W_IN + 2 * wo;
    const float2 top = *reinterpret_cast<const float2*>(&smem[base]);          // row 2r
    const float2 bot = *reinterpret_cast<const float2*>(&smem[base + W_IN]);   // row 2r+1
    w0[k] = top.x; w1[k] = top.y; w2[k] = bot.x; w3[k] = bot.y;
  }

  float m0 = 0.f, m1 = 0.f, m2 = 0.f, m3 = 0.f;
#pragma unroll
  for (int k = 0; k < 4; ++k) {
    m0 = fmaf(w0[k], w0[k], m0);
    m1 = fmaf(w1[k], w1[k], m1);
    m2 = fmaf(w2[k], w2[k], m2);
    m3 = fmaf(w3[k], w3[k], m3);
  }

  // First-occurrence argmax over p = 0..3 (row-major window order).
  int   bi   = 0;
  float best = m0;
  if (m1 > best) { best = m1; bi = 1; }
  if (m2 > best) { best = m2; bi = 2; }
  if (m3 > best) { best = m3; bi = 3; }

  // ---- Store: out[g*4+k][hb*4+r][wo], coalesced b32 per component plane. ----
  float* obase = out + (size_t)g * QUAD_OUT + (size_t)(hb * 4 + r) * WO + wo;
#pragma unroll
  for (int k = 0; k < 4; ++k) {
    float v = w0[k];
    if (bi == 1) v = w1[k];
    if (bi == 2) v = w2[k];
    if (bi == 3) v = w3[k];
    obase[(size_t)k * OUT_PLANE] = v;
  }
}

extern "C" void kernel_launch(void* const* d_in, const int* in_sizes, int n_in,
                              void* d_out, int out_size, void* d_ws, size_t ws_size,
                              hipStream_t stream) {
  (void)in_sizes; (void)n_in; (void)d_ws; (void)ws_size; (void)out_size;
  const float* x = (const float*)d_in[0];
  float* out = (float*)d_out;
  // 1024 quads * 16 row-blocks = 16384 blocks of 256 threads.
  ga2_maxpool_kernel<<<dim3(G_TOTAL * HB_PER_G), dim3(256), 0, stream>>>(x, out);
}